// DeeppredLayers_51299089383663
// MI455X (gfx1250) — compile-verified
//
#include <hip/hip_runtime.h>
#include <hip/hip_bf16.h>

typedef __attribute__((ext_vector_type(2))) float v2f;
typedef __attribute__((ext_vector_type(8))) float v8f;

#define DP_MAXN 8192

// ---------------------------------------------------------------------------
// CDNA5 async global->LDS copy (16B per lane), tracked by ASYNCcnt.
// lds_off is the byte offset within the workgroup's LDS allocation
// (low 32 bits of a flat LDS pointer).
// ---------------------------------------------------------------------------
__device__ __forceinline__ void async_g2l_b128(unsigned lds_off,
                                               const float* gaddr) {
    asm volatile("global_load_async_to_lds_b128 %0, %1, off"
                 :: "v"(lds_off), "v"(gaddr)
                 : "memory");
}

__device__ __forceinline__ void wait_async_done() {
    asm volatile("s_wait_asynccnt 0x0" ::: "memory");
}

// ---------------------------------------------------------------------------
// Zero the two scalar sum accumulators (mean computation) each launch.
// ---------------------------------------------------------------------------
__global__ void dp_init_kernel(float* __restrict__ sums) {
    if (threadIdx.x < 2) sums[threadIdx.x] = 0.0f;
}

// ---------------------------------------------------------------------------
// Per-wave WMMA matvec over a K-range.
//   y[rowbase + m] += sum_k W[rowbase+m][k] * x[k],   k in [k0, k0+kspan)
// A-tile (16x4 f32): lane L<16 holds row M=L, K={k,k+1}; lane L>=16 holds
// row M=L-16, K={k+2,k+3}  -> one global_load_b64 per lane per WMMA (pure
// weight stream on the VMEM pipe).
// B (4x16): B[k][n] = x[k] for all n (broadcast from LDS via ds_load_b64),
// so every column of D accumulates the same 16 partial dot products.
// ---------------------------------------------------------------------------
__device__ __forceinline__ v8f wmma_matvec_range(const float* __restrict__ W,
                                                 const float* __restrict__ xlds,
                                                 int rowbase, int N,
                                                 int k0, int kspan, int lane) {
    const int half = lane >> 4;      // 0: K={0,1}, 1: K={2,3}
    const int r    = lane & 15;      // matrix row within 16-row tile
    const float* a_ptr = W + (size_t)(rowbase + r) * (size_t)N + (size_t)(k0 + 2 * half);
    const float* b_ptr = xlds + (size_t)(k0 + 2 * half);

    v8f acc0 = {};
    v8f acc1 = {};
    #pragma unroll 4
    for (int k = 0; k < kspan; k += 8) {
        v2f a0 = *(const v2f*)(a_ptr);
        v2f b0 = *(const v2f*)(b_ptr);
        v2f a1 = *(const v2f*)(a_ptr + 4);
        v2f b1 = *(const v2f*)(b_ptr + 4);
        acc0 = __builtin_amdgcn_wmma_f32_16x16x4_f32(false, a0, false, b0,
                                                     (short)0, acc0, false, false);
        acc1 = __builtin_amdgcn_wmma_f32_16x16x4_f32(false, a1, false, b1,
                                                     (short)0, acc1, false, false);
        a_ptr += 8;
        b_ptr += 8;
    }
    return acc0 + acc1;
}

// ---------------------------------------------------------------------------
// Fused matvec kernel.
// Blocks [0, N/16):      layer_in = lower_act@W_layer.T + lower_p@W_lower_p.T
//                        (waves 0-3: W_layer 4-way K-split; waves 4-7: W_lower_p)
// Blocks [N/16, 2*N/16): p_in = higher_ct@W_ct_to_p.T (8-way K-split)
// x vectors are staged to LDS via async global->LDS copies at block start.
// Each block owns 16 output rows; LDS float-atomics reduce across waves;
// one global float-atomic per block accumulates the vector sum for the mean.
// ---------------------------------------------------------------------------
__global__ void __launch_bounds__(256)
dp_matvec_kernel(const float* __restrict__ lower_act,
                 const float* __restrict__ lower_p,
                 const float* __restrict__ higher_ct,
                 const float* __restrict__ W_layer,
                 const float* __restrict__ W_lower_p,
                 const float* __restrict__ W_ct_to_p,
                 float* __restrict__ vec,   // [0,N): layer_in, [N,2N): p_in
                 float* __restrict__ sums,  // [0]: sum(layer_in), [1]: sum(p_in)
                 int N) {
    __shared__ float s_x[2 * DP_MAXN];       // 64 KB vector staging
    __shared__ float s_out[16];

    const int half_blocks = (int)gridDim.x >> 1;           // N/16
    const bool isP        = (int)blockIdx.x >= half_blocks;
    const int  rowbase    = ((int)blockIdx.x - (isP ? half_blocks : 0)) * 16;
    const int  wave       = threadIdx.x >> 5;              // wave32
    const int  lane       = threadIdx.x & 31;

    // ---- Stage x vector(s) into LDS with async copies (ASYNCcnt path) ----
    {
        const float* v0 = isP ? higher_ct : lower_act;
        const unsigned base0 = (unsigned)(uintptr_t)(&s_x[0]);
        const unsigned base1 = (unsigned)(uintptr_t)(&s_x[N]);
        for (int off = (int)threadIdx.x * 4; off < N; off += 256 * 4) {
            async_g2l_b128(base0 + (unsigned)off * 4u, v0 + off);
            if (!isP) {
                async_g2l_b128(base1 + (unsigned)off * 4u, lower_p + off);
            }
        }
        wait_async_done();
    }
    if (threadIdx.x < 16) s_out[threadIdx.x] = 0.0f;
    __syncthreads();

    const float* W;
    const float* xlds;
    int k0, kspan;
    if (isP) {
        W     = W_ct_to_p;
        xlds  = s_x;
        kspan = N >> 3;                                    // 8-way K split
        k0    = wave * kspan;
    } else {
        const bool second = wave >= 4;
        W     = second ? W_lower_p : W_layer;
        xlds  = second ? (s_x + N) : s_x;
        kspan = N >> 2;                                    // 4-way K split
        k0    = (wave & 3) * kspan;
    }

    v8f acc = wmma_matvec_range(W, xlds, rowbase, N, k0, kspan, lane);

    // D layout: acc[v] holds y[rowbase + v] in lanes 0-15 and
    // y[rowbase + 8 + v] in lanes 16-31 (all columns identical).
    if (lane == 0 || lane == 16) {
        const int base = (lane >> 4) * 8;
        #pragma unroll
        for (int i = 0; i < 8; ++i) {
            atomicAdd(&s_out[base + i], acc[i]);
        }
    }
    __syncthreads();

    float* dst = vec + (isP ? N : 0) + rowbase;
    if (threadIdx.x < 16) dst[threadIdx.x] = s_out[threadIdx.x];

    if (threadIdx.x == 0) {
        float s = 0.0f;
        #pragma unroll
        for (int i = 0; i < 16; ++i) s += s_out[i];
        atomicAdd(&sums[isP ? 1 : 0], s);
    }
}

// ---------------------------------------------------------------------------
// Activation: ffi = max(mean, 0); out = tanh(max(in - ffi, 0) * 8)
// ---------------------------------------------------------------------------
__global__ void __launch_bounds__(256)
dp_act_kernel(const float* __restrict__ vec,
              const float* __restrict__ sums,
              float* __restrict__ out,
              int N) {
    const int i = (int)blockIdx.x * (int)blockDim.x + (int)threadIdx.x;
    if (i >= 2 * N) return;
    const int which = (i >= N) ? 1 : 0;
    const float inv = vec[i];                  // vec already laid out [layer | p]
    const float ffi = fmaxf(sums[which] / (float)N, 0.0f);
    out[i] = tanhf(fmaxf(inv - ffi, 0.0f) * 8.0f);
}

// ---------------------------------------------------------------------------
extern "C" void kernel_launch(void* const* d_in, const int* in_sizes, int n_in,
                              void* d_out, int out_size, void* d_ws, size_t ws_size,
                              hipStream_t stream) {
    const float* lower_act = (const float*)d_in[0];
    const float* lower_p   = (const float*)d_in[1];
    const float* higher_ct = (const float*)d_in[2];
    const float* W_layer   = (const float*)d_in[3];
    const float* W_lower_p = (const float*)d_in[4];
    const float* W_ct_to_p = (const float*)d_in[5];

    const int N = in_sizes[0];                 // 8192

    float* vec  = (float*)d_ws;                // 2N floats: layer_in | p_in
    float* sums = vec + 2 * (size_t)N;         // 2 floats

    dp_init_kernel<<<1, 32, 0, stream>>>(sums);

    const int blocks = 2 * (N / 16);           // 1024 blocks of 16 rows each
    dp_matvec_kernel<<<blocks, 256, 0, stream>>>(
        lower_act, lower_p, higher_ct,
        W_layer, W_lower_p, W_ct_to_p,
        vec, sums, N);

    const int act_threads = 2 * N;
    dp_act_kernel<<<(act_threads + 255) / 256, 256, 0, stream>>>(
        vec, sums, (float*)d_out, N);
}